// PLPConv_19765439496456
// MI455X (gfx1250) — compile-verified
//
#include <hip/hip_runtime.h>
#include <cstdint>

// PLPConv: out = D^-1/2 (A+I) D^-1/2 H   (N=100k, C=48, E=1.6M)
// MI455X reasoning: compute is ~160 MFLOP (irrelevant); label/out/dis are
// L2-resident (192MB L2); bottleneck is 76.8M f32 L2 atomics + 25.6MB edge
// stream from HBM. WMMA is inapplicable (random sparse scatter). CDNA5 path
// used: async global->LDS edge staging (ASYNCcnt, double buffered).

#define CCH   48
#define CCH4  12        // float4 chunks per row
#define TILE  256       // edges per tile (== blockDim)

__global__ void init_deg_kernel(float* __restrict__ deg, int N) {
  int i = blockIdx.x * blockDim.x + threadIdx.x;
  if (i < N) deg[i] = 1.0f;           // self-loop contributes 1 to every degree
}

__global__ void degree_kernel(const long long* __restrict__ row,
                              float* __restrict__ deg, int E) {
  int i = blockIdx.x * blockDim.x + threadIdx.x;
  int stride = gridDim.x * blockDim.x;
  for (; i < E; i += stride) {
    int r = (int)row[i];
    atomicAdd(&deg[r], 1.0f);
  }
}

__global__ void rsqrt_kernel(const float* __restrict__ deg,
                             float* __restrict__ dis, int N) {
  int i = blockIdx.x * blockDim.x + threadIdx.x;
  if (i < N) dis[i] = rsqrtf(deg[i]); // deg >= 1 always, no zero case
}

// out[n] = dis[n]^2 * label[n]  -- plain stores, fully initializes d_out
__global__ void selfloop_kernel(const float4* __restrict__ label4,
                                const float* __restrict__ dis,
                                float4* __restrict__ out4, int M /* = N*CCH4 */) {
  int i = blockIdx.x * blockDim.x + threadIdx.x;
  if (i >= M) return;
  int n = i / CCH4;
  float d = dis[n];
  float s = d * d;
  float4 v = label4[i];
  v.x *= s; v.y *= s; v.z *= s; v.w *= s;
  out4[i] = v;
}

// Edge scatter. Each block loops over TILE-edge tiles; the int64 row/col
// indices of the NEXT tile are streamed into LDS with
// global_load_async_to_lds_b128 (ASYNCcnt) while the CURRENT tile's
// gather/scale/atomic-scatter runs. One async b128 per thread per tile
// (128 threads cover 2KB of rows, 128 cover 2KB of cols).
__global__ void scatter_kernel(const float4* __restrict__ label4,
                               const long long* __restrict__ edge, // rows [0,E), cols [E,2E)
                               const float* __restrict__ dis,
                               float* __restrict__ out,
                               int E, int ntiles) {
  __shared__ long long buf[2][2 * TILE]; // [0..255]=row, [256..511]=col

  const int tid = threadIdx.x;
  const uint64_t base = (uint64_t)edge;            // wave-uniform saddr
  const long long colShift = (tid < 128) ? 0ll : (long long)E * 8; // wave-uniform per wave
  const int lane16 = (tid & 127) * 16;             // byte offset inside tile's 2KB half
  const long long maxoff = (long long)E * 16 - 16; // clamp: stay inside edge_index buffer
  const int slot = (tid < 128) ? ((tid & 127) * 2) : (TILE + (tid & 127) * 2);

  int t = blockIdx.x;
  int cur = 0;

  if (t < ntiles) {
    long long off = colShift + (long long)t * (TILE * 8) + lane16;
    if (off > maxoff) off = maxoff;
    unsigned lds = (unsigned)(uintptr_t)&buf[0][slot];
    asm volatile("global_load_async_to_lds_b128 %0, %1, %2"
                 :: "v"(lds), "v"((unsigned)off), "s"(base) : "memory");
  }

  for (; t < ntiles; t += gridDim.x) {
    int tn = t + gridDim.x;
    if (tn < ntiles) {
      long long off = colShift + (long long)tn * (TILE * 8) + lane16;
      if (off > maxoff) off = maxoff;
      unsigned lds = (unsigned)(uintptr_t)&buf[cur ^ 1][slot];
      asm volatile("global_load_async_to_lds_b128 %0, %1, %2"
                   :: "v"(lds), "v"((unsigned)off), "s"(base) : "memory");
      asm volatile("s_wait_asynccnt 0x1" ::: "memory"); // tile t done, t+stride in flight
    } else {
      asm volatile("s_wait_asynccnt 0x0" ::: "memory");
    }
    __syncthreads(); // all waves' async data visible in LDS

    int e = t * TILE + tid;
    if (e < E) {
      int r = (int)buf[cur][tid];
      int c = (int)buf[cur][TILE + tid];
      float norm = dis[r] * dis[c];           // L2-resident gathers
      const float4* lp = label4 + (size_t)r * CCH4;
      float* op = out + (size_t)c * CCH;
#pragma unroll
      for (int k = 0; k < CCH4; ++k) {
        float4 v = lp[k];                     // global_load_b128, L2 hit
        atomicAdd(op + 4 * k + 0, v.x * norm);
        atomicAdd(op + 4 * k + 1, v.y * norm);
        atomicAdd(op + 4 * k + 2, v.z * norm);
        atomicAdd(op + 4 * k + 3, v.w * norm);
      }
    }
    __syncthreads(); // done reading buf[cur] before it is refilled next iter
    cur ^= 1;
  }
}

extern "C" void kernel_launch(void* const* d_in, const int* in_sizes, int n_in,
                              void* d_out, int out_size, void* d_ws, size_t ws_size,
                              hipStream_t stream) {
  const float* label = (const float*)d_in[0];
  const long long* edge = (const long long*)d_in[1]; // int64 (2,E) flat: rows then cols

  const int N = in_sizes[0] / CCH;
  const int E = in_sizes[1] / 2;
  float* out = (float*)d_out;
  float* deg = (float*)d_ws;       // N floats
  float* dis = deg + N;            // N floats

  const int B = 256;

  init_deg_kernel<<<(N + B - 1) / B, B, 0, stream>>>(deg, N);

  degree_kernel<<<2048, B, 0, stream>>>(edge, deg, E);

  rsqrt_kernel<<<(N + B - 1) / B, B, 0, stream>>>(deg, dis, N);

  const int M = N * CCH4;
  selfloop_kernel<<<(M + B - 1) / B, B, 0, stream>>>(
      (const float4*)label, dis, (float4*)out, M);

  const int ntiles = (E + TILE - 1) / TILE;
  const int grid = ntiles < 2048 ? ntiles : 2048;
  scatter_kernel<<<grid, TILE, 0, stream>>>(
      (const float4*)label, edge, dis, out, E, ntiles);
}